// PointQuantizer_33646773796930
// MI455X (gfx1250) — compile-verified
//
#include <hip/hip_runtime.h>
#include <hip/hip_bf16.h>

typedef __attribute__((ext_vector_type(2))) float v2f;
typedef __attribute__((ext_vector_type(8))) float v8f;

// ---------------------------------------------------------------------------
// Kernel 1: brute-force nearest-codebook-entry via V_WMMA_F32_16X16X4_F32.
// Each wave owns 16 points (rows). Loop over K/16 tiles of 16 grid points.
//   A (16x4, MxK): lanes 0-15 hold {x,y} (K=0,1) of point M=lane,
//                  lanes 16-31 hold {z,0} (K=2,3) of point M=lane-16.
//   B (4x16, KxN): lanes 0-15 hold {cx,cy} of grid col N=lane,
//                  lanes 16-31 hold {cz,0} of grid col N=lane-16.
//   D (16x16 f32): VGPR g -> row M=g (lanes 0-15) / M=g+8 (lanes 16-31),
//                  column N = lane & 15.
// Score = |c|^2 - 2*(x.c); the per-row |x|^2 constant cannot change argmin.
// The grid-tile load is software-pipelined one tile deep so the VMEM latency
// overlaps the WMMA + argmin chain instead of stalling on s_wait_loadcnt.
// ---------------------------------------------------------------------------
__global__ __launch_bounds__(256)
void pq_argmin_wmma(const float* __restrict__ x,
                    const float* __restrict__ gp,
                    int* __restrict__ enc,
                    int K) {
    const int wave  = threadIdx.x >> 5;
    const int lane  = threadIdx.x & 31;
    const int half  = lane >> 4;          // 0: K=0,1 side   1: K=2,3 side
    const int l16   = lane & 15;
    const int mBase = (blockIdx.x * 8 + wave) * 16;

    // A operand (constant across the K loop)
    const int m = mBase + l16;
    v2f a;
    if (half == 0) { a.x = x[m * 3 + 0]; a.y = x[m * 3 + 1]; }
    else           { a.x = x[m * 3 + 2]; a.y = 0.0f;         }

    float best[8];
    int   bidx[8];
#pragma unroll
    for (int g = 0; g < 8; ++g) { best[g] = __builtin_inff(); bidx[g] = 0; }

    const int mask = K - 1;               // K is a power of two (32768)

    // prologue: fetch tile 0
    float cx = gp[l16 * 3 + 0];
    float cy = gp[l16 * 3 + 1];
    float cz = gp[l16 * 3 + 2];

#pragma unroll 2
    for (int kt = 0; kt < K; kt += 16) {
        // prefetch the NEXT tile (wraps to tile 0 on the last trip; harmless)
        const int nn = ((kt + 16) & mask) + l16;
        const float nx = gp[nn * 3 + 0];
        const float ny = gp[nn * 3 + 1];
        const float nz = gp[nn * 3 + 2];

        const int n = kt + l16;                         // current grid column
        const float cn = cx * cx + cy * cy + cz * cz;   // |c_n|^2

        v2f b;
        if (half == 0) { b.x = cx; b.y = cy; }
        else           { b.x = cz; b.y = 0.0f; }

        v8f d = {};
        d = __builtin_amdgcn_wmma_f32_16x16x4_f32(
                /*neg_a=*/false, a, /*neg_b=*/false, b,
                /*c_mod=*/(short)0, d, /*reuse_a=*/false, /*reuse_b=*/false);

#pragma unroll
        for (int g = 0; g < 8; ++g) {
            const float s = cn - 2.0f * d[g];
            // strict < keeps the first (lowest-index) minimum, since n grows
            if (s < best[g]) { best[g] = s; bidx[g] = n; }
        }

        cx = nx; cy = ny; cz = nz;        // rotate double buffer
    }

    // Cross-lane argmin within each 16-lane half (the lanes sharing a row set).
#pragma unroll
    for (int off = 8; off >= 1; off >>= 1) {
#pragma unroll
        for (int g = 0; g < 8; ++g) {
            const float ob = __shfl_xor(best[g], off, 16);
            const int   oi = __shfl_xor(bidx[g], off, 16);
            if (ob < best[g] || (ob == best[g] && oi < bidx[g])) {
                best[g] = ob; bidx[g] = oi;
            }
        }
    }

    if (l16 == 0) {
#pragma unroll
        for (int g = 0; g < 8; ++g)
            enc[mBase + half * 8 + g] = bidx[g];   // rows g / g+8 per half
    }
}

// ---------------------------------------------------------------------------
// Kernel 2: bitonic sort of one 1024-entry batch row in LDS; emit as float.
// ---------------------------------------------------------------------------
__global__ __launch_bounds__(1024)
void pq_sort_row(const int* __restrict__ enc, float* __restrict__ out) {
    __shared__ int s[1024];
    const int t = threadIdx.x;
    s[t] = enc[blockIdx.x * 1024 + t];
    __syncthreads();

    for (int k = 2; k <= 1024; k <<= 1) {
        for (int j = k >> 1; j > 0; j >>= 1) {
            const int ixj = t ^ j;
            if (ixj > t) {
                const int va = s[t], vb = s[ixj];
                const bool up = ((t & k) == 0);
                if ((va > vb) == up) { s[t] = vb; s[ixj] = va; }
            }
            __syncthreads();
        }
    }
    out[blockIdx.x * 1024 + t] = (float)s[t];
}

// ---------------------------------------------------------------------------
// Kernel 3: fill voxel region with -0.8 ( (0 - 0.5)/0.5 * 0.8 ).
// ---------------------------------------------------------------------------
__global__ __launch_bounds__(256)
void pq_voxel_fill(float* __restrict__ vox, int n) {
    const int i = blockIdx.x * blockDim.x + threadIdx.x;
    if (i < n) vox[i] = -0.8f;
}

// ---------------------------------------------------------------------------
// Kernel 4: scatter +0.8 at occupied voxels (duplicates write same value).
// ---------------------------------------------------------------------------
__global__ __launch_bounds__(256)
void pq_voxel_scatter(const int* __restrict__ enc, float* __restrict__ vox,
                      int K, int npts) {
    const int p = blockIdx.x * blockDim.x + threadIdx.x;
    if (p < npts) {
        const int b = p >> 10;            // 1024 points per batch row
        vox[b * K + enc[p]] = 0.8f;
    }
}

extern "C" void kernel_launch(void* const* d_in, const int* in_sizes, int n_in,
                              void* d_out, int out_size, void* d_ws, size_t ws_size,
                              hipStream_t stream) {
    const float* x  = (const float*)d_in[0];   // (B, N, 3) f32
    const float* gp = (const float*)d_in[1];   // (K, 3)    f32
    float* out = (float*)d_out;                // [enc (B*N) | voxel (B*K)] as f32

    const int npts = in_sizes[0] / 3;          // 4096
    const int K    = in_sizes[1] / 3;          // 32768
    const int B    = npts / 1024;

    int* enc = (int*)d_ws;                     // npts int32 scratch

    // 1) WMMA argmin: 256 row-tiles of 16 points, 8 waves per 256-thread block.
    pq_argmin_wmma<<<(npts / 16) / 8, 256, 0, stream>>>(x, gp, enc, K);

    // 2) per-row bitonic sort -> enc region of d_out (as float, exact <= 32767)
    pq_sort_row<<<B, 1024, 0, stream>>>(enc, out);

    // 3) + 4) voxel occupancy
    const int voxN = B * K;
    pq_voxel_fill<<<(voxN + 255) / 256, 256, 0, stream>>>(out + npts, voxN);
    pq_voxel_scatter<<<(npts + 255) / 256, 256, 0, stream>>>(enc, out + npts, K, npts);
}